// NCEAverage_pcl_8229157339417
// MI455X (gfx1250) — compile-verified
//
#include <hip/hip_runtime.h>

// ---------------------------------------------------------------------------
// NCEAverage-style contrastive scoring for MI455X (gfx1250, wave32, WMMA).
//
// Stage 1: feat_proj = l2norm(feat @ W^T + b)            -> d_ws
// Stage 2: scores[r] = dot(l2norm(memory[idx[r]] @ W^T + b), feat_proj[b]) / T
// Stage 3: new_memory = memory, rows y <- 0.5*memory[y] + 0.5*feat
//
// Big GEMM (65536x512x2048, 137 GFLOP fp32) on V_WMMA_F32_16X16X4_F32.
// 32-row tiles: each W (B) fragment feeds 2 WMMAs -> vmem:wmma = 1:2.
// A rows staged via GLOBAL_LOAD_ASYNC_TO_LDS_B128 (ASYNCcnt), no VGPR bounce.
// Dominant cost: 1.34 GB new_memory stream (~58 us @ 23.3 TB/s).
// ---------------------------------------------------------------------------

typedef float v2f __attribute__((ext_vector_type(2)));
typedef float v8f __attribute__((ext_vector_type(8)));

#define D_IN    2048
#define D_PROJ  512
#define KP1     1024
#define BATCH   64
#define NMEM    81920
#define KC      64            // K-chunk staged in LDS
#define RT      32            // rows per workgroup tile (2 x 16-row WMMA subtiles)
#define INV_T   (1.0f / 0.07f)

typedef __attribute__((address_space(3))) float lds_f32;

__device__ __forceinline__ unsigned lds_offset(const void* p) {
    // generic -> LDS addrspace cast yields the raw 32-bit LDS byte offset
    return (unsigned)(unsigned long long)(lds_f32*)p;
}

__device__ __forceinline__ void async_load_b128_to_lds(unsigned lds_off,
                                                       unsigned long long gaddr) {
    asm volatile("global_load_async_to_lds_b128 %0, %1, off"
                 :: "v"(lds_off), "v"(gaddr) : "memory");
}

__device__ __forceinline__ void wait_asynccnt0() {
    asm volatile("s_wait_asynccnt 0x0" ::: "memory");
}

// mode 0: src rows are feat[row];          out = normalized projections [rows x 512]
// mode 1: src rows are memory[idx[row]];   out = scores [rows]
__global__ __launch_bounds__(256)
void proj_wmma_kernel(const float* __restrict__ src,
                      const int*   __restrict__ idx,
                      const float* __restrict__ W,      // [512 x 2048] row-major
                      const float* __restrict__ bias,   // [512]
                      const float* __restrict__ fp,     // featproj (mode 1 only)
                      float*       __restrict__ out,
                      int mode)
{
    __shared__ float A_lds[RT][KC];          // gathered A chunk (32 rows x 64 K) = 8 KB
    __shared__ float fp_lds[D_PROJ];         // feat_proj row for this tile's batch
    __shared__ float bias_lds[D_PROJ];
    __shared__ float red_sq[RT];
    __shared__ float red_dot[RT];

    const int tid  = threadIdx.x;
    const int wave = tid >> 5;
    const int lane = tid & 31;
    const int half = lane >> 4;              // ISA A/B layout: lanes 16-31 hold K+2,K+3
    const int l16  = lane & 15;
    const int row0 = blockIdx.x * RT;        // 32-row tile; 1024 % 32 == 0 -> single batch
    const int bb   = row0 / KP1;

    for (int c = tid; c < D_PROJ; c += 256) {
        bias_lds[c] = bias[c];
        if (mode == 1) fp_lds[c] = fp[bb * D_PROJ + c];
    }
    if (tid < RT) { red_sq[tid] = 0.f; red_dot[tid] = 0.f; }

    // this thread's two staging assignments (32 rows x 16 float4 slots = 512)
    const int r0 = tid >> 4;                 // rows 0..15
    const int r1 = (tid + 256) >> 4;         // rows 16..31
    const int c4 = (tid & 15) * 4;
    long srow0, srow1;
    if (mode == 1) { srow0 = (long)idx[row0 + r0]; srow1 = (long)idx[row0 + r1]; }
    else           { srow0 = (long)(row0 + r0);    srow1 = (long)(row0 + r1);    }
    const unsigned ldsa0 = lds_offset(&A_lds[r0][c4]);
    const unsigned ldsa1 = lds_offset(&A_lds[r1][c4]);

    v8f acc[2][4];
#pragma unroll
    for (int s = 0; s < 2; ++s)
#pragma unroll
        for (int ct = 0; ct < 4; ++ct)
            acc[s][ct] = (v8f){0.f,0.f,0.f,0.f,0.f,0.f,0.f,0.f};

    const int waveCol0 = wave * 64;          // 8 waves x 64 cols = 512 cols

    for (int k0 = 0; k0 < D_IN; k0 += KC) {
        // async-stage A chunk straight into LDS (no VGPR round trip)
        async_load_b128_to_lds(ldsa0,
            (unsigned long long)(src + srow0 * (long)D_IN + k0 + c4));
        async_load_b128_to_lds(ldsa1,
            (unsigned long long)(src + srow1 * (long)D_IN + k0 + c4));
        wait_asynccnt0();
        __syncthreads();

        // 16 K-steps of V_WMMA_F32_16X16X4_F32; each B fragment feeds 2 subtiles
        for (int ks = 0; ks < KC; ks += 4) {
            v2f a0, a1;                       // A 16x4: lane<16 -> K0,K1 ; lane>=16 -> K2,K3
            a0.x = A_lds[l16][ks + 2 * half + 0];
            a0.y = A_lds[l16][ks + 2 * half + 1];
            a1.x = A_lds[16 + l16][ks + 2 * half + 0];
            a1.y = A_lds[16 + l16][ks + 2 * half + 1];
#pragma unroll
            for (int ct = 0; ct < 4; ++ct) {
                const int n = waveCol0 + ct * 16 + l16;   // B 4x16: N = lane&15
                const float* wrow = W + (long)n * D_IN + k0 + ks + 2 * half;
                v2f b;
                b.x = wrow[0];
                b.y = wrow[1];
                acc[0][ct] = __builtin_amdgcn_wmma_f32_16x16x4_f32(
                    false, a0, false, b, (short)0, acc[0][ct], false, false);
                acc[1][ct] = __builtin_amdgcn_wmma_f32_16x16x4_f32(
                    false, a1, false, b, (short)0, acc[1][ct], false, false);
            }
        }
        __syncthreads();
    }

    // epilogue straight from accumulators.
    // C/D layout: VGPR j holds M = j + 8*half, N = lane&15 (per column tile).
#pragma unroll
    for (int s = 0; s < 2; ++s) {
#pragma unroll
        for (int j = 0; j < 8; ++j) {
            float sq = 0.f, dt = 0.f;
#pragma unroll
            for (int ct = 0; ct < 4; ++ct) {
                const int n = waveCol0 + ct * 16 + l16;
                const float v = acc[s][ct][j] + bias_lds[n];
                sq += v * v;
                if (mode == 1) dt += v * fp_lds[n];
            }
            const int m = s * 16 + j + 8 * half;
            atomicAdd(&red_sq[m], sq);                  // ds_add_f32
            if (mode == 1) atomicAdd(&red_dot[m], dt);
        }
    }
    __syncthreads();

    if (mode == 0) {
        // write normalized projection directly from registers
#pragma unroll
        for (int s = 0; s < 2; ++s) {
#pragma unroll
            for (int j = 0; j < 8; ++j) {
                const int m = s * 16 + j + 8 * half;
                const float inv = 1.0f / sqrtf(red_sq[m]);
#pragma unroll
                for (int ct = 0; ct < 4; ++ct) {
                    const int n = waveCol0 + ct * 16 + l16;
                    const float v = acc[s][ct][j] + bias_lds[n];
                    out[(long)(row0 + m) * D_PROJ + n] = v * inv;
                }
            }
        }
    } else if (tid < RT) {
        out[row0 + tid] = red_dot[tid] / sqrtf(red_sq[tid]) * INV_T;
    }
}

// new_memory = memory  (671 MB read + 671 MB write: the HBM roofline term)
__global__ __launch_bounds__(256)
void copy_mem_kernel(const float4* __restrict__ src, float4* __restrict__ dst, long n4)
{
    long i = (long)blockIdx.x * blockDim.x + threadIdx.x;
    const long stride = (long)gridDim.x * blockDim.x;
    for (; i < n4; i += stride) dst[i] = src[i];
}

// momentum update of rows y (last-occurrence wins to match .at[y].set semantics)
__global__ __launch_bounds__(256)
void update_rows_kernel(const float* __restrict__ feat,
                        const int*   __restrict__ y,
                        const float* __restrict__ memory,
                        float*       __restrict__ dst)
{
    const int b  = blockIdx.x;
    const int yb = y[b];
    for (int bp = b + 1; bp < BATCH; ++bp)
        if (y[bp] == yb) return;             // a later write to this row wins
    const long base = (long)yb * D_IN;
    for (int c = threadIdx.x; c < D_IN; c += blockDim.x)
        dst[base + c] = 0.5f * memory[base + c] + 0.5f * feat[(long)b * D_IN + c];
}

extern "C" void kernel_launch(void* const* d_in, const int* in_sizes, int n_in,
                              void* d_out, int out_size, void* d_ws, size_t ws_size,
                              hipStream_t stream)
{
    const float* feat   = (const float*)d_in[0];   // [64, 2048]
    const int*   y      = (const int*)  d_in[1];   // [64]
    const int*   idx    = (const int*)  d_in[2];   // [64, 1024]
    const float* memory = (const float*)d_in[3];   // [81920, 2048]
    const float* W      = (const float*)d_in[4];   // [512, 2048]
    const float* bias   = (const float*)d_in[5];   // [512]

    float* out_scores = (float*)d_out;                               // [64*1024]
    float* out_mem    = (float*)d_out + (size_t)BATCH * KP1;         // [81920*2048]
    float* fp         = (float*)d_ws;                                // [64*512]

    // Stage 1: feat_proj (2 tiles of 32 rows)
    proj_wmma_kernel<<<BATCH / RT, 256, 0, stream>>>(
        feat, nullptr, W, bias, nullptr, fp, 0);

    // Stage 2: gathered projection + fused normalized dot (2048 tiles)
    proj_wmma_kernel<<<(BATCH * KP1) / RT, 256, 0, stream>>>(
        memory, idx, W, bias, fp, out_scores, 1);

    // Stage 3: stream the memory bank into the output, then patch the y rows
    const long n4 = (long)NMEM * D_IN / 4;
    copy_mem_kernel<<<16384, 256, 0, stream>>>(
        (const float4*)memory, (float4*)out_mem, n4);
    update_rows_kernel<<<BATCH, 256, 0, stream>>>(feat, y, memory, out_mem);
}